// MinusSpan_15384572854730
// MI455X (gfx1250) — compile-verified
//
#include <hip/hip_runtime.h>
#include <cstdint>
#include <cstddef>

// Problem constants (match reference: B=16, T=2048, D=1024, N_SPANS=256)
constexpr int kB  = 16;
constexpr int kT  = 2048;
constexpr int kD  = 1024;
constexpr int kH  = 512;    // D/2
constexpr int kNS = 256;

typedef float f4 __attribute__((ext_vector_type(4)));

#if defined(__gfx1250__)
#define SPAN_ASYNC 1
#else
#define SPAN_ASYNC 0
#endif

#if SPAN_ASYNC
// Fire-and-forget 16B global -> LDS DMA (CDNA5 async path, ASYNCcnt).
// vdst operand = 32-bit LDS byte address, vaddr = full 64-bit global address.
__device__ __forceinline__ void async_ld_b128(const float* g, uint32_t lds_byte) {
    uint64_t ga = (uint64_t)(uintptr_t)g;
    asm volatile("global_load_async_to_lds_b128 %0, %1, off"
                 :: "v"(lds_byte), "v"(ga)
                 : "memory");
}
__device__ __forceinline__ void wait_async0() {
    asm volatile("s_wait_asynccnt 0" ::: "memory");
}
#endif

__global__ __launch_bounds__(128)
void span_gather_kernel(const float* __restrict__ inp,
                        const int*   __restrict__ spans,
                        float*       __restrict__ out)
{
    const int span = blockIdx.x;        // 0 .. B*NS-1
    const int b    = span >> 8;         // span / kNS
    const int t    = threadIdx.x;       // 0 .. 127

    const int i = spans[2 * span + 0];
    const int j = spans[2 * span + 1];

    const bool skip  = (i == 0) && (j == 0);
    const bool prev  = (i >= 1);
    const bool postv = (j + 1 < kT);

    const float* base = inp + (size_t)b * kT * kD;
    float*       o    = out + (size_t)span * (2 * kD);

    const int c = t * 4;                // float index inside a 512-float row

    const f4 vzero = {0.f, 0.f, 0.f, 0.f};
    f4 fe, bs, fp, bp;

#if SPAN_ASYNC
    // LDS staging: 4 rows x 512 floats = 8 KB per workgroup.
    __shared__ float lds[4 * kH];
    const uint32_t l_fend = (uint32_t)(uintptr_t)&lds[0 * kH + c];
    const uint32_t l_fpre = (uint32_t)(uintptr_t)&lds[1 * kH + c];
    const uint32_t l_bst  = (uint32_t)(uintptr_t)&lds[2 * kH + c];
    const uint32_t l_bpo  = (uint32_t)(uintptr_t)&lds[3 * kH + c];

    // i, j are uniform per block -> these branches are wave-uniform (EXEC full).
    async_ld_b128(base + (size_t)j * kD + c,            l_fend);
    async_ld_b128(base + (size_t)i * kD + kH + c,       l_bst);
    if (prev)  async_ld_b128(base + (size_t)(i - 1) * kD + c,      l_fpre);
    if (postv) async_ld_b128(base + (size_t)(j + 1) * kD + kH + c, l_bpo);
    wait_async0();

    fe = *(const f4*)&lds[0 * kH + c];
    bs = *(const f4*)&lds[2 * kH + c];
    fp = prev  ? *(const f4*)&lds[1 * kH + c] : vzero;
    bp = postv ? *(const f4*)&lds[3 * kH + c] : vzero;
#else
    fe = *(const f4*)(base + (size_t)j * kD + c);
    bs = *(const f4*)(base + (size_t)i * kD + kH + c);
    fp = prev  ? *(const f4*)(base + (size_t)(i - 1) * kD + c)      : vzero;
    bp = postv ? *(const f4*)(base + (size_t)(j + 1) * kD + kH + c) : vzero;
#endif

    if (skip) { fe = vzero; bs = vzero; fp = vzero; bp = vzero; }

    // out = [f_end - f_pre | b_start - b_post | f_pre | b_post]
    *(f4*)(o + 0 * kH + c) = fe - fp;
    *(f4*)(o + 1 * kH + c) = bs - bp;
    *(f4*)(o + 2 * kH + c) = fp;
    *(f4*)(o + 3 * kH + c) = bp;
}

extern "C" void kernel_launch(void* const* d_in, const int* in_sizes, int n_in,
                              void* d_out, int out_size, void* d_ws, size_t ws_size,
                              hipStream_t stream) {
    (void)in_sizes; (void)n_in; (void)out_size; (void)d_ws; (void)ws_size;

    const float* inp   = (const float*)d_in[0];
    const int*   spans = (const int*)d_in[1];   // (B, NS, 2) int
    float*       out   = (float*)d_out;         // (B, NS, 2*D) float

    dim3 grid(kB * kNS);   // 4096 blocks, one span each
    dim3 block(128);       // 4 waves; thread t owns 16B chunk t of each segment
    hipLaunchKernelGGL(span_gather_kernel, grid, block, 0, stream,
                       inp, spans, out);
}